// FPSKNNGrouper_15771119911461
// MI455X (gfx1250) — compile-verified
//
#include <hip/hip_runtime.h>

#define NB      8
#define NPOINT  16384
#define NSAMP   512
#define KNNK    16
#define NCH     16

typedef float v2f __attribute__((ext_vector_type(2)));
typedef float v8f __attribute__((ext_vector_type(8)));

// ---------------------------------------------------------------------------
// Kernel 1: furthest point sampling. One 1024-thread workgroup per batch.
// Each thread owns 16 points (coords + running min-dist in registers).
// Writes sampled (x,y,z,|s|^2) as float4 to workspace.
// ---------------------------------------------------------------------------
__global__ __launch_bounds__(1024) void fps_kernel(const float* __restrict__ x,
                                                   float4* __restrict__ samp) {
  const int b   = blockIdx.x;
  const int tid = threadIdx.x;              // 0..1023
  const float* xb = x + (size_t)b * NPOINT * NCH;

  float px[16], py[16], pz[16], md[16];
#pragma unroll
  for (int i = 0; i < 16; ++i) {
    const int p = tid * 16 + i;
    const float4 q = *(const float4*)(xb + (size_t)p * NCH);
    px[i] = q.x; py[i] = q.y; pz[i] = q.z;
    md[i] = 1e10f;                          // matches reference init 10000000000.0
  }

  __shared__ float s_val[32];
  __shared__ int   s_idx[32];
  __shared__ float s_last[4];

  if (tid == 0) {
    const float4 q = *(const float4*)xb;    // seed = point 0
    s_last[0] = q.x; s_last[1] = q.y; s_last[2] = q.z;
    samp[b * NSAMP + 0] = make_float4(q.x, q.y, q.z, q.x*q.x + q.y*q.y + q.z*q.z);
  }
  __syncthreads();

  for (int t = 1; t < NSAMP; ++t) {
    const float lx = s_last[0], ly = s_last[1], lz = s_last[2];
    float bestv = -1.0f;
    int   besti = NPOINT;
#pragma unroll
    for (int i = 0; i < 16; ++i) {
      const float dx = px[i] - lx, dy = py[i] - ly, dz = pz[i] - lz;
      const float d  = dx*dx + dy*dy + dz*dz;
      md[i] = fminf(md[i], d);
      if (md[i] > bestv) { bestv = md[i]; besti = tid * 16 + i; }  // strict > keeps lowest i
    }
    // wave32 argmax reduce, ties -> lower index (jnp.argmax semantics)
#pragma unroll
    for (int off = 16; off > 0; off >>= 1) {
      const float ov = __shfl_down(bestv, off, 32);
      const int   oi = __shfl_down(besti, off, 32);
      if (ov > bestv || (ov == bestv && oi < besti)) { bestv = ov; besti = oi; }
    }
    if ((tid & 31) == 0) { s_val[tid >> 5] = bestv; s_idx[tid >> 5] = besti; }
    __syncthreads();
    if (tid < 32) {
      bestv = s_val[tid]; besti = s_idx[tid];
#pragma unroll
      for (int off = 16; off > 0; off >>= 1) {
        const float ov = __shfl_down(bestv, off, 32);
        const int   oi = __shfl_down(besti, off, 32);
        if (ov > bestv || (ov == bestv && oi < besti)) { bestv = ov; besti = oi; }
      }
      if (tid == 0) {
        const float4 q = *(const float4*)(xb + (size_t)besti * NCH);
        s_last[0] = q.x; s_last[1] = q.y; s_last[2] = q.z;
        samp[b * NSAMP + t] = make_float4(q.x, q.y, q.z, q.x*q.x + q.y*q.y + q.z*q.z);
      }
    }
    __syncthreads();
  }
}

// ---------------------------------------------------------------------------
// Kernel 2: WMMA distance tiles, LDS-resident per-lane top-16 lists (column
// per lane), register-cached guard, runtime bubble insert, LDS merge, gather.
// One wave per (batch, 16-sample tile). grid = (32 tiles, 8 batches).
// ---------------------------------------------------------------------------
__global__ __launch_bounds__(32) void knn_kernel(const float* __restrict__ x,
                                                 const float4* __restrict__ samp,
                                                 float* __restrict__ out) {
  const int b    = blockIdx.y;
  const int tile = blockIdx.x;              // 0..31
  const int lane = threadIdx.x;             // 0..31
  const int nl   = lane & 15;
  const int hi   = lane >> 4;
  const float* xb = x + (size_t)b * NPOINT * NCH;

  __shared__ float s_dot[256];              // 16x16 dot tile
  __shared__ float s_p2[32];                // full |p|^2 per column (0..15 used)
  __shared__ float s_bd[KNNK][32];          // per-lane sorted dist list (column/lane)
  __shared__ int   s_bi[KNNK][32];
  __shared__ int   s_knn[256];              // merged neighbor indices

  // A-matrix (16x4 f32): lane row m = nl; V0 = K(2*hi), V1 = K(2*hi+1); K=3 padded 0.
  const float4 sp = samp[b * NSAMP + tile * 16 + nl];
  v2f A;
  A.x = hi ? sp.z : sp.x;
  A.y = hi ? 0.0f : sp.y;
  const float s2 = sp.w;                    // |s|^2 for row m = nl
  const int   m  = nl;                      // sample row owned by this lane

  // init per-lane list (own column, no barrier needed before own use)
#pragma unroll 1
  for (int k = 0; k < KNNK; ++k) { s_bd[k][lane] = 3.0e38f; s_bi[k][lane] = 0x7fffffff; }
  float worst  = 3.0e38f;                   // register cache of s_bd[15][lane]
  int   worsti = 0x7fffffff;

  for (int j = 0; j < NPOINT; j += 16) {
    // prefetch next candidate chunk (clamped in-bounds) -> global_prefetch_b8
    const int jn = (j + 16 < NPOINT) ? (j + 16) : j;
    __builtin_prefetch(xb + (size_t)(jn + nl) * NCH, 0, 0);

    // B-matrix (4x16 f32): col n = nl; V0 = K(2*hi), V1 = K(2*hi+1); K=3 padded 0.
    const float2 q = *(const float2*)(xb + (size_t)(j + nl) * NCH + 2 * hi);
    v2f Bv;
    Bv.x = q.x;
    Bv.y = hi ? 0.0f : q.y;
    // full |p|^2 for column nl: combine the two half-wave partials via shfl_xor
    const float part = Bv.x * Bv.x + Bv.y * Bv.y;
    s_p2[lane] = part + __shfl_xor(part, 16, 32);

    v8f C = {};
    C = __builtin_amdgcn_wmma_f32_16x16x4_f32(false, A, false, Bv,
                                              (short)0, C, false, false);
#pragma unroll
    for (int r = 0; r < 8; ++r)
      s_dot[(r + 8 * hi) * 16 + nl] = C[r]; // D: VGPR r -> (M = r + 8*hi, N = nl)
    __syncthreads();

    // all 32 lanes scan: lane handles row m = nl, columns n = 8*hi .. 8*hi+7
#pragma unroll
    for (int nn = 0; nn < 8; ++nn) {
      const int   n   = 8 * hi + nn;
      const float d   = (s2 + s_p2[n]) - 2.0f * s_dot[m * 16 + n];
      const int   idx = j + n;
      // lexicographic (d, idx): reproduces stable argsort order
      if (d < worst || (d == worst && idx < worsti)) {
        s_bd[KNNK - 1][lane] = d; s_bi[KNNK - 1][lane] = idx;
#pragma unroll 1
        for (int t2 = KNNK - 1; t2 > 0; --t2) {
          const float ud = s_bd[t2 - 1][lane];
          const int   ui = s_bi[t2 - 1][lane];
          const bool  sw = (d < ud) || (d == ud && idx < ui);
          if (!sw) break;
          s_bd[t2][lane] = ud;      s_bi[t2][lane] = ui;
          s_bd[t2 - 1][lane] = d;   s_bi[t2 - 1][lane] = idx;
        }
        worst  = s_bd[KNNK - 1][lane];
        worsti = s_bi[KNNK - 1][lane];
      }
    }
    __syncthreads();
  }

  // lanes 0..15: merge the two sorted 16-lists for row m (stable lexicographic).
  // List A = lane m's column, list B = lane (m+16)'s column.
  if (hi == 0) {
    int pa = 0, pb = 0;
#pragma unroll 1
    for (int k = 0; k < KNNK; ++k) {
      const bool va = (pa < KNNK), vb = (pb < KNNK);
      const float da = va ? s_bd[va ? pa : 0][m]      : __builtin_inff();
      const int   ia = va ? s_bi[va ? pa : 0][m]      : 0x7fffffff;
      const float db = vb ? s_bd[vb ? pb : 0][m + 16] : __builtin_inff();
      const int   ib = vb ? s_bi[vb ? pb : 0][m + 16] : 0x7fffffff;
      const bool ta = (da < db) || (da == db && ia < ib);
      s_knn[m * KNNK + k] = ta ? ia : ib;
      pa += ta ? 1 : 0;
      pb += ta ? 0 : 1;
    }
  }
  __syncthreads();

  // Gather output: [b][tile*16+m][k][0..15], 256 rows of 16 floats per block.
  float* ob = out + ((size_t)b * NSAMP + tile * 16) * KNNK * NCH;
  for (int rk = lane; rk < 16 * KNNK; rk += 32) {
    const int mm = rk >> 4, k = rk & 15;
    const int idx = s_knn[mm * KNNK + k];
    const float4* src = (const float4*)(xb + (size_t)idx * NCH);
    float4* dst = (float4*)(ob + (size_t)(mm * KNNK + k) * NCH);
    dst[0] = src[0]; dst[1] = src[1]; dst[2] = src[2]; dst[3] = src[3];
  }
}

// ---------------------------------------------------------------------------
extern "C" void kernel_launch(void* const* d_in, const int* in_sizes, int n_in,
                              void* d_out, int out_size, void* d_ws, size_t ws_size,
                              hipStream_t stream) {
  const float* x   = (const float*)d_in[0];        // [8,16384,16] f32
  float*       out = (float*)d_out;                // [8,512,16,16] f32
  float4*      smp = (float4*)d_ws;                // 8*512 float4 = 64 KB scratch

  fps_kernel<<<NB, 1024, 0, stream>>>(x, smp);
  knn_kernel<<<dim3(NSAMP / 16, NB), 32, 0, stream>>>(x, smp, out);
}